// MultiHorizonDistHead_6743098655427
// MI455X (gfx1250) — compile-verified
//
#include <hip/hip_runtime.h>
#include <cfloat>
#include <cmath>

typedef float v2f __attribute__((ext_vector_type(2)));
typedef float v8f __attribute__((ext_vector_type(8)));

namespace {

constexpr int Bsz = 1024, Lsz = 512, Dsz = 512, Fsz = 64;
constexpr int Hn = 3, Kn = 4, Qn = 5;
constexpr int SYM = 16, REG = 16, DTEXT = 64, DQn = 32;
constexpr int CTXK = Dsz + SYM + DTEXT + DQn; // 624

// output offsets (floats, concatenated in reference return order)
constexpr int O_MU   = 0;
constexpr int O_LS   = Bsz * Hn;             // 3072
constexpr int O_QO   = 2 * Bsz * Hn;         // 6144
constexpr int O_DIR  = O_QO + Bsz * Hn * Qn; // 21504
constexpr int O_GATE = O_DIR + Bsz * Hn;     // 24576
constexpr int O_RP   = O_GATE + Bsz * Kn;    // 28672
constexpr int O_DF   = O_RP + Bsz * 3;       // 31744

// workspace offsets (floats)
constexpr size_t W_TM   = 0;                                   // B*L
constexpr size_t W_PC   = (size_t)Bsz * Lsz;                   // pooled_ctx 3072x624
constexpr size_t W_MIX  = W_PC + (size_t)Bsz * Hn * CTXK;      // mixed 3072x512
constexpr size_t W_GATE = W_MIX + (size_t)Bsz * Hn * Dsz;      // B*4
constexpr size_t W_REG  = W_GATE + (size_t)Bsz * Kn;           // B*16

__device__ __forceinline__ float wave_sum(float v) {
  v += __shfl_xor(v, 16, 32);
  v += __shfl_xor(v, 8, 32);
  v += __shfl_xor(v, 4, 32);
  v += __shfl_xor(v, 2, 32);
  v += __shfl_xor(v, 1, 32);
  return v;
}
__device__ __forceinline__ float wave_max(float v) {
  v = fmaxf(v, __shfl_xor(v, 16, 32));
  v = fmaxf(v, __shfl_xor(v, 8, 32));
  v = fmaxf(v, __shfl_xor(v, 4, 32));
  v = fmaxf(v, __shfl_xor(v, 2, 32));
  v = fmaxf(v, __shfl_xor(v, 1, 32));
  return v;
}
__device__ __forceinline__ float block_sum_256(float v, float* red) {
  const int t = threadIdx.x;
  red[t] = v; __syncthreads();
  if (t < 128) red[t] += red[t + 128];
  __syncthreads();
  if (t < 64) red[t] += red[t + 64];
  __syncthreads();
  if (t < 32) {
    float x = red[t] + red[t + 32];
    x = wave_sum(x);
    if (t == 0) red[0] = x;
  }
  __syncthreads();
  float r = red[0];
  __syncthreads();
  return r;
}
__device__ __forceinline__ float block_max_256(float v, float* red) {
  const int t = threadIdx.x;
  red[t] = v; __syncthreads();
  if (t < 128) red[t] = fmaxf(red[t], red[t + 128]);
  __syncthreads();
  if (t < 64) red[t] = fmaxf(red[t], red[t + 64]);
  __syncthreads();
  if (t < 32) {
    float x = fmaxf(red[t], red[t + 32]);
    x = wave_max(x);
    if (t == 0) red[0] = x;
  }
  __syncthreads();
  float r = red[0];
  __syncthreads();
  return r;
}
__device__ __forceinline__ float gelu_exact(float x) {
  return 0.5f * x * (1.f + erff(x * 0.70710678118654752440f));
}

// ---------------------------------------------------------------------------
// K1: mask statistics + small MLPs (text/qual/router/regime classifier)
// ---------------------------------------------------------------------------
__global__ __launch_bounds__(256) void k1_features(
    const float* __restrict__ x_values, const float* __restrict__ x_mask,
    const float* __restrict__ static_ctx,
    const float* __restrict__ regime_f, const float* __restrict__ regime_m,
    const float* __restrict__ text_W, const float* __restrict__ text_b,
    const float* __restrict__ qual_W, const float* __restrict__ qual_b,
    const float* __restrict__ r1_W, const float* __restrict__ r1_b,
    const float* __restrict__ r2_W, const float* __restrict__ r2_b,
    const float* __restrict__ rc1_W, const float* __restrict__ rc1_b,
    const float* __restrict__ rc2_W, const float* __restrict__ rc2_b,
    float* __restrict__ tm_ws, float* __restrict__ pooled_ctx,
    float* __restrict__ gate_ws, float* __restrict__ regime_ws,
    float* __restrict__ out) {
  __shared__ float sm_pnum[8][24];
  __shared__ float sm_pden[8][24];
  __shared__ float sm_feat[24];   // [0:16)=tfeat, [16:24)=qfeat
  __shared__ float sm_text[64];
  __shared__ float sm_qvec[32];
  __shared__ float sm_regime[16];
  __shared__ float sm_static[16];
  __shared__ float sm_hid[64];
  __shared__ float sm_rh[16];
  __shared__ float sm_l4[4];
  __shared__ float sm_l3[3];

  const int b = blockIdx.x;
  const int tid = threadIdx.x;
  const int lane = tid & 31;
  const int wave = tid >> 5;

  // pass over L: time_missing + masked sums for features 40..63
  float num = 0.f, den = 0.f;
  for (int l = wave; l < Lsz; l += 8) {
    const size_t base = ((size_t)b * Lsz + l) * Fsz;
    float m1 = x_mask[base + lane];
    float m2 = x_mask[base + lane + 32];
    float s = wave_sum(m1 + m2);
    if (lane == 0)
      tm_ws[(size_t)b * Lsz + l] = (s * (1.f / 64.f) >= 0.999f) ? 1.f : 0.f;
    if (lane >= 8) {  // feature f = lane+32 in [40,64)
      float w = fminf(fmaxf(1.f - m2, 0.f), 1.f);
      den += w;
      num += w * x_values[base + lane + 32];
    }
  }
  if (lane >= 8) { sm_pnum[wave][lane - 8] = num; sm_pden[wave][lane - 8] = den; }
  __syncthreads();

  if (tid < 24) {
    float n = 0.f, d = 0.f;
    for (int w = 0; w < 8; ++w) { n += sm_pnum[w][tid]; d += sm_pden[w][tid]; }
    sm_feat[tid] = n / fmaxf(d, 1e-8f);
  }
  if (tid >= 96 && tid < 112) {
    int i = tid - 96;
    float rg = regime_f[b * REG + i] *
               fminf(fmaxf(1.f - regime_m[b * REG + i], 0.f), 1.f);
    sm_regime[i] = rg;
    regime_ws[(size_t)b * REG + i] = rg;
  }
  if (tid >= 112 && tid < 128) sm_static[tid - 112] = static_ctx[b * SYM + tid - 112];
  __syncthreads();

  if (tid < 64) {
    float a = text_b[tid];
    for (int i = 0; i < 16; ++i) a += sm_feat[i] * text_W[i * 64 + tid];
    sm_text[tid] = tanhf(a);
  } else if (tid < 96) {
    int j = tid - 64;
    float a = qual_b[j];
    for (int i = 0; i < 8; ++i) a += sm_feat[16 + i] * qual_W[i * 32 + j];
    sm_qvec[j] = tanhf(a);
  }
  __syncthreads();

  if (tid < 64) {  // router hidden: in = [regime(16), static(16), q_vec(32)]
    float a = r1_b[tid];
    for (int i = 0; i < 64; ++i) {
      float x = (i < 16) ? sm_regime[i] : (i < 32) ? sm_static[i - 16] : sm_qvec[i - 32];
      a += x * r1_W[i * 64 + tid];
    }
    sm_hid[tid] = gelu_exact(a);
  } else if (tid < 80) {  // regime classifier hidden
    int j = tid - 64;
    float a = rc1_b[j];
    for (int i = 0; i < 16; ++i) a += sm_regime[i] * rc1_W[i * 16 + j];
    sm_rh[j] = gelu_exact(a);
  }
  __syncthreads();

  if (tid < 4) {
    float a = r2_b[tid];
    for (int j = 0; j < 64; ++j) a += sm_hid[j] * r2_W[j * 4 + tid];
    sm_l4[tid] = a;
  } else if (tid < 7) {
    int k = tid - 4;
    float a = rc2_b[k];
    for (int j = 0; j < 16; ++j) a += sm_rh[j] * rc2_W[j * 3 + k];
    sm_l3[k] = a;
  }
  __syncthreads();

  if (tid == 0) {  // softmax(4) -> top-2 -> normalized sparse gate
    float mx = fmaxf(fmaxf(sm_l4[0], sm_l4[1]), fmaxf(sm_l4[2], sm_l4[3]));
    float e[4], s = 0.f;
    for (int i = 0; i < 4; ++i) { e[i] = expf(sm_l4[i] - mx); s += e[i]; }
    float p[4];
    for (int i = 0; i < 4; ++i) p[i] = e[i] / s;
    int i1 = 0;
    for (int i = 1; i < 4; ++i) if (p[i] > p[i1]) i1 = i;
    int i2 = (i1 == 0) ? 1 : 0;
    for (int i = 0; i < 4; ++i) if (i != i1 && p[i] > p[i2]) i2 = i;
    float dn = fmaxf(p[i1] + p[i2], 1e-8f);
    float g[4] = {0.f, 0.f, 0.f, 0.f};
    g[i1] = p[i1] / dn; g[i2] = p[i2] / dn;
    for (int i = 0; i < 4; ++i) {
      out[O_GATE + b * 4 + i] = g[i];
      gate_ws[(size_t)b * 4 + i] = g[i];
    }
  } else if (tid == 1) {
    float mx = fmaxf(fmaxf(sm_l3[0], sm_l3[1]), sm_l3[2]);
    float e0 = expf(sm_l3[0] - mx), e1 = expf(sm_l3[1] - mx), e2 = expf(sm_l3[2] - mx);
    float inv = 1.f / (e0 + e1 + e2);
    out[O_RP + b * 3 + 0] = e0 * inv;
    out[O_RP + b * 3 + 1] = e1 * inv;
    out[O_RP + b * 3 + 2] = e2 * inv;
  }
  if (tid < 112) {  // pooled_ctx tail columns [512,624): static|text|qvec
    float v = (tid < 16) ? sm_static[tid] : (tid < 80) ? sm_text[tid - 16] : sm_qvec[tid - 80];
    for (int h = 0; h < Hn; ++h)
      pooled_ctx[((size_t)(b * Hn + h)) * CTXK + Dsz + tid] = v;
  }
}

// ---------------------------------------------------------------------------
// K2: attention scores -> masked softmax -> pooled -> layernorm
// ---------------------------------------------------------------------------
struct K2SM {
  alignas(16) float q[Hn * Dsz];   // 1536
  alignas(16) float sc[Hn * Lsz];  // 1536: scores then attn
  float tm[Lsz];
  float red[256];
};

__global__ __launch_bounds__(256) void k2_attention(
    const float* __restrict__ h_seq, const float* __restrict__ query,
    const float* __restrict__ ln_g, const float* __restrict__ ln_b,
    const float* __restrict__ tm_ws, float* __restrict__ pooled_ctx) {
  __shared__ K2SM sm;
  const int b = blockIdx.x;
  const int tid = threadIdx.x;
  const int lane = tid & 31;
  const int wave = tid >> 5;

  for (int i = tid; i < Hn * Dsz; i += 256) sm.q[i] = query[i];
  for (int i = tid; i < Lsz; i += 256) sm.tm[i] = tm_ws[(size_t)b * Lsz + i];
  __syncthreads();

  // pass 1: scores[h][l] — one wave per row l, coalesced float4 stream
  for (int l = wave; l < Lsz; l += 8) {
    const float* hp = h_seq + ((size_t)b * Lsz + l) * Dsz;
    float a0 = 0.f, a1 = 0.f, a2 = 0.f;
#pragma unroll
    for (int it = 0; it < 4; ++it) {
      int d = it * 128 + lane * 4;
      float4 hv = *(const float4*)(hp + d);
      float4 q0 = *(const float4*)(&sm.q[d]);
      float4 q1 = *(const float4*)(&sm.q[Dsz + d]);
      float4 q2 = *(const float4*)(&sm.q[2 * Dsz + d]);
      a0 += hv.x * q0.x + hv.y * q0.y + hv.z * q0.z + hv.w * q0.w;
      a1 += hv.x * q1.x + hv.y * q1.y + hv.z * q1.z + hv.w * q1.w;
      a2 += hv.x * q2.x + hv.y * q2.y + hv.z * q2.z + hv.w * q2.w;
    }
    a0 = wave_sum(a0); a1 = wave_sum(a1); a2 = wave_sum(a2);
    if (lane == 0) { sm.sc[l] = a0; sm.sc[Lsz + l] = a1; sm.sc[2 * Lsz + l] = a2; }
  }
  __syncthreads();

  // masked softmax over L per head (all-masked row -> all zeros)
  float tm0 = sm.tm[tid], tm1 = sm.tm[tid + 256];
  for (int h = 0; h < Hn; ++h) {
    float s0 = sm.sc[h * Lsz + tid], s1 = sm.sc[h * Lsz + tid + 256];
    float m0 = (tm0 > 0.f) ? -FLT_MAX : s0;
    float m1 = (tm1 > 0.f) ? -FLT_MAX : s1;
    float mx = block_max_256(fmaxf(m0, m1), sm.red);
    float e0 = (m0 == -FLT_MAX) ? 0.f : expf(m0 - mx);
    float e1 = (m1 == -FLT_MAX) ? 0.f : expf(m1 - mx);
    float s = block_sum_256(e0 + e1, sm.red);
    float inv = (s > 0.f) ? 1.f / s : 0.f;
    sm.sc[h * Lsz + tid] = e0 * inv;
    sm.sc[h * Lsz + tid + 256] = e1 * inv;
    __syncthreads();
  }

  // pass 2: pooled[h][d] = sum_l attn[h][l] * h_seq[l][d] (second, L2-hot pass)
  float p00 = 0, p01 = 0, p10 = 0, p11 = 0, p20 = 0, p21 = 0;
  const int d0 = tid, d1 = tid + 256;
  for (int l = 0; l < Lsz; ++l) {
    const float* hp = h_seq + ((size_t)b * Lsz + l) * Dsz;
    float h0 = hp[d0], h1 = hp[d1];
    float a0 = sm.sc[l], a1 = sm.sc[Lsz + l], a2 = sm.sc[2 * Lsz + l];
    p00 += a0 * h0; p01 += a0 * h1;
    p10 += a1 * h0; p11 += a1 * h1;
    p20 += a2 * h0; p21 += a2 * h1;
  }

  float g0v = ln_g[d0], g1v = ln_g[d1], b0v = ln_b[d0], b1v = ln_b[d1];
  float ph[3][2] = {{p00, p01}, {p10, p11}, {p20, p21}};
  for (int h = 0; h < Hn; ++h) {
    float tot = block_sum_256(ph[h][0] + ph[h][1], sm.red);
    float mean = tot * (1.f / 512.f);
    float c0 = ph[h][0] - mean, c1 = ph[h][1] - mean;
    float vtot = block_sum_256(c0 * c0 + c1 * c1, sm.red);
    float inv = rsqrtf(vtot * (1.f / 512.f) + 1e-5f);
    size_t row = ((size_t)(b * Hn + h)) * CTXK;
    pooled_ctx[row + d0] = c0 * inv * g0v + b0v;
    pooled_ctx[row + d1] = c1 * inv * g1v + b1v;
  }
}

// ---------------------------------------------------------------------------
// K3: mixed = tanh(pooled_ctx @ ctx_W + ctx_b) via V_WMMA_F32_16X16X4_F32
//   A: 3072 x 624 row-major, W: 624 x 512 row-major, C: 3072 x 512.
//   Block = 8 waves; block owns 16 rows; wave owns 64 cols (4 16x16 tiles).
//   A frag 16x4: lane&15 = M row, lanes>=16 hold K+2 (x=K, y=K+1).
//   B frag 4x16: lane&15 = N col, lanes>=16 hold K+2 rows.
//   C/D frag: VGPR v holds row v (lanes 0-15) / row v+8 (lanes 16-31).
// ---------------------------------------------------------------------------
__global__ __launch_bounds__(256) void k3_ctx_gemm(
    const float* __restrict__ A, const float* __restrict__ Wm,
    const float* __restrict__ bias, float* __restrict__ Cm) {
  const int lane = threadIdx.x & 31;
  const int wave = threadIdx.x >> 5;
  const int m0 = blockIdx.x << 4;
  const int nb = wave << 6;
  const int mrow = m0 + (lane & 15);
  const int koff = (lane >> 4) << 1;
  const int ncol = nb + (lane & 15);

  v8f acc[4] = {v8f{}, v8f{}, v8f{}, v8f{}};
  const float* Arow = A + (size_t)mrow * CTXK;

  for (int kk = 0; kk < CTXK; kk += 4) {
    v2f a;
    a.x = Arow[kk + koff];
    a.y = Arow[kk + koff + 1];
    const float* W0 = Wm + (size_t)(kk + koff) * Dsz;
    const float* W1 = W0 + Dsz;
    v2f b0; b0.x = W0[ncol];      b0.y = W1[ncol];
    v2f b1; b1.x = W0[ncol + 16]; b1.y = W1[ncol + 16];
    v2f b2; b2.x = W0[ncol + 32]; b2.y = W1[ncol + 32];
    v2f b3; b3.x = W0[ncol + 48]; b3.y = W1[ncol + 48];
    acc[0] = __builtin_amdgcn_wmma_f32_16x16x4_f32(false, a, false, b0, (short)0, acc[0], false, false);
    acc[1] = __builtin_amdgcn_wmma_f32_16x16x4_f32(false, a, false, b1, (short)0, acc[1], false, false);
    acc[2] = __builtin_amdgcn_wmma_f32_16x16x4_f32(false, a, false, b2, (short)0, acc[2], false, false);
    acc[3] = __builtin_amdgcn_wmma_f32_16x16x4_f32(false, a, false, b3, (short)0, acc[3], false, false);
  }

  const int rbase = m0 + ((lane >> 4) << 3);
#pragma unroll
  for (int j = 0; j < 4; ++j) {
    const int col = nb + j * 16 + (lane & 15);
    const float bv = bias[col];
#pragma unroll
    for (int v = 0; v < 8; ++v) {
      const int row = rbase + v;
      Cm[(size_t)row * Dsz + col] = tanhf(acc[j][v] + bv);
    }
  }
}

// ---------------------------------------------------------------------------
// K4: expert dots + gate mixture + sigma/stress + quantile sort + df
// ---------------------------------------------------------------------------
__global__ __launch_bounds__(256) void k4_head(
    const float* __restrict__ mixed,
    const float* __restrict__ mu_W, const float* __restrict__ mu_b,
    const float* __restrict__ ls_W, const float* __restrict__ ls_b,
    const float* __restrict__ dir_W, const float* __restrict__ dir_b,
    const float* __restrict__ q_Wm, const float* __restrict__ q_b,
    const float* __restrict__ df_W, const float* __restrict__ df_b,
    const float* __restrict__ gate_ws, const float* __restrict__ regime_ws,
    float* __restrict__ out) {
  __shared__ float s_mix[Hn * Dsz];  // 1536
  __shared__ float s_dot[99];

  const int b = blockIdx.x;
  const int tid = threadIdx.x;
  const int lane = tid & 31;
  const int wave = tid >> 5;

  for (int i = tid; i < Hn * Dsz; i += 256)
    s_mix[i] = mixed[(size_t)b * Hn * Dsz + i];
  __syncthreads();

  // 99 dots: per h (33): k in 0..3 x {mu,ls,dir,q0..q4} (8 each) + df
  for (int t = wave; t < 99; t += 8) {
    const int h = t / 33, r = t % 33;
    const float* wp;
    int stride = 1, doclip = 0;
    float bias;
    if (r == 32) {
      wp = df_W; bias = df_b[0];
    } else {
      const int k = r >> 3, w = r & 7;
      if (w == 0)      { wp = mu_W + k * Dsz;  bias = mu_b[k]; }
      else if (w == 1) { wp = ls_W + k * Dsz;  bias = ls_b[k]; doclip = 1; }
      else if (w == 2) { wp = dir_W + k * Dsz; bias = dir_b[k]; }
      else {
        const int q = w - 3;
        wp = q_Wm + (size_t)k * Dsz * Qn + q;
        stride = Qn;
        bias = q_b[k * Qn + q];
      }
    }
    float acc = 0.f;
    for (int i = lane; i < Dsz; i += 32) acc += s_mix[h * Dsz + i] * wp[i * stride];
    acc = wave_sum(acc);
    if (lane == 0) {
      float v = acc + bias;
      if (doclip) v = fminf(fmaxf(v, -7.f), 2.f);
      s_dot[t] = v;
    }
  }
  __syncthreads();

  if (tid < Hn) {
    const int h = tid;
    float g[4];
    for (int k = 0; k < 4; ++k) g[k] = gate_ws[(size_t)b * 4 + k];
    float mu = 0.f, ls = 0.f, dr = 0.f, qd[5] = {0.f, 0.f, 0.f, 0.f, 0.f};
    for (int k = 0; k < 4; ++k) {
      const int base = h * 33 + k * 8;
      mu += g[k] * s_dot[base + 0];
      ls += g[k] * s_dot[base + 1];
      dr += g[k] * s_dot[base + 2];
      for (int q = 0; q < 5; ++q) qd[q] += g[k] * s_dot[base + 3 + q];
    }
    float sigma = fmaxf(expf(ls), 1e-6f);
    const float r0 = regime_ws[(size_t)b * REG + 0];
    const float r3 = regime_ws[(size_t)b * REG + 3];
    const float r10 = regime_ws[(size_t)b * REG + 10];
    const float stress = fmaxf(0.f, r0 + fmaxf(r3, 0.f) + fmaxf(-r10, 0.f));
    sigma = sigma * (1.f + 0.25f * g[2] * stress);
    const float lso = logf(fmaxf(sigma, 1e-6f));

    float v[5];
    for (int q = 0; q < 5; ++q) v[q] = mu + sigma * qd[q];
    for (int i = 1; i < 5; ++i) {  // insertion sort ascending
      float x = v[i];
      int j = i - 1;
      while (j >= 0 && v[j] > x) { v[j + 1] = v[j]; --j; }
      v[j + 1] = x;
    }
    const float dd = s_dot[h * 33 + 32];
    const float df = 2.f + ((dd > 20.f) ? dd : log1pf(expf(dd)));

    const int bh = b * Hn + h;
    out[O_MU + bh] = mu;
    out[O_LS + bh] = lso;
    out[O_DIR + bh] = dr;
    out[O_DF + bh] = df;
    for (int q = 0; q < 5; ++q) out[O_QO + bh * Qn + q] = v[q];
  }
}

}  // namespace

extern "C" void kernel_launch(void* const* d_in, const int* in_sizes, int n_in,
                              void* d_out, int out_size, void* d_ws, size_t ws_size,
                              hipStream_t stream) {
  (void)in_sizes; (void)n_in; (void)out_size; (void)ws_size;
  const float* h_seq      = (const float*)d_in[0];
  const float* x_values   = (const float*)d_in[1];
  const float* x_mask     = (const float*)d_in[2];
  const float* static_ctx = (const float*)d_in[3];
  const float* regime_f   = (const float*)d_in[4];
  const float* regime_m   = (const float*)d_in[5];
  const float* query      = (const float*)d_in[6];
  const float* ln_g       = (const float*)d_in[7];
  const float* ln_b       = (const float*)d_in[8];
  const float* ctx_W      = (const float*)d_in[9];
  const float* ctx_b      = (const float*)d_in[10];
  const float* text_W     = (const float*)d_in[11];
  const float* text_b     = (const float*)d_in[12];
  const float* qual_W     = (const float*)d_in[13];
  const float* qual_b     = (const float*)d_in[14];
  const float* r1_W       = (const float*)d_in[15];
  const float* r1_b       = (const float*)d_in[16];
  const float* r2_W       = (const float*)d_in[17];
  const float* r2_b       = (const float*)d_in[18];
  const float* rc1_W      = (const float*)d_in[19];
  const float* rc1_b      = (const float*)d_in[20];
  const float* rc2_W      = (const float*)d_in[21];
  const float* rc2_b      = (const float*)d_in[22];
  const float* mu_W       = (const float*)d_in[23];
  const float* mu_b       = (const float*)d_in[24];
  const float* ls_W       = (const float*)d_in[25];
  const float* ls_b       = (const float*)d_in[26];
  const float* dir_W      = (const float*)d_in[27];
  const float* dir_b      = (const float*)d_in[28];
  const float* q_W        = (const float*)d_in[29];
  const float* q_b        = (const float*)d_in[30];
  const float* df_W       = (const float*)d_in[31];
  const float* df_b       = (const float*)d_in[32];

  float* ws = (float*)d_ws;
  float* out = (float*)d_out;
  float* tm  = ws + W_TM;
  float* pc  = ws + W_PC;
  float* mix = ws + W_MIX;
  float* gws = ws + W_GATE;
  float* rws = ws + W_REG;

  k1_features<<<Bsz, 256, 0, stream>>>(
      x_values, x_mask, static_ctx, regime_f, regime_m,
      text_W, text_b, qual_W, qual_b, r1_W, r1_b, r2_W, r2_b,
      rc1_W, rc1_b, rc2_W, rc2_b, tm, pc, gws, rws, out);

  k2_attention<<<Bsz, 256, 0, stream>>>(h_seq, query, ln_g, ln_b, tm, pc);

  k3_ctx_gemm<<<(Bsz * Hn) / 16, 256, 0, stream>>>(pc, ctx_W, ctx_b, mix);

  k4_head<<<Bsz, 256, 0, stream>>>(
      mix, mu_W, mu_b, ls_W, ls_b, dir_W, dir_b, q_W, q_b, df_W, df_b,
      gws, rws, out);
}